// estimate_light_47914655154717
// MI455X (gfx1250) — compile-verified
//
#include <hip/hip_runtime.h>

// CDNA5 / gfx1250: wave32, WMMA f32 16x16x4 for the Gram-matrix accumulation.

typedef __attribute__((ext_vector_type(2))) float v2f;
typedef __attribute__((ext_vector_type(8))) float v8f;

#define NBATCH 8
#define H_     512
#define W_     512
#define HW_    (H_ * W_)
#define BLOCKS_PER_BATCH 64
#define PIX_PER_BLOCK    (HW_ / BLOCKS_PER_BATCH)   // 4096
#define ITERS            (PIX_PER_BLOCK / 256)      // 16
#define LDS_STRIDE 17   // 16 + 1 pad: 17 coprime with 64 banks -> conflict-free column reads

// ---------------------------------------------------------------------------
// Kernel 1: fused SH-basis + masked scaling + 16x16 Gram accumulation (WMMA).
// Each block: 256 threads = 8 waves, handles 4096 pixels of one batch image.
// Per iteration each thread computes one pixel's 16-vector
//   G = [A0..A8, b0, b1, b2, 0,0,0,0]
// into LDS; each wave then runs 8x V_WMMA_F32_16X16X4_F32 over its private
// 32x16 sub-tile, chaining the f32 accumulator:  C += Gt * G  (16x16).
//   - AtA  = C[0:9, 0:9]
//   - Atb  = C[9+c, 0:9]   (c = RGB channel)
// A-fragment layout (ISA 7.12.2, 32-bit A 16x4): lane(0..15)=M, v0=K0|K2,
// v1=K1|K3. With the mirrored B layout the Gram's A and B fragments carry
// identical register data, so one v2f feeds both operands.
// ---------------------------------------------------------------------------
__global__ __launch_bounds__(256)
void sh_gram_kernel(const float* __restrict__ input_img,
                    const float* __restrict__ depth_target,
                    const float* __restrict__ albedo,
                    float* __restrict__ partial)   // [NBATCH][BLOCKS_PER_BATCH][256]
{
    __shared__ float sh[256 * LDS_STRIDE];
    __shared__ float red[8 * 256];

    const int tid  = threadIdx.x;
    const int lane = tid & 31;
    const int wave = tid >> 5;
    const int half = lane >> 4;     // 0: lanes 0-15, 1: lanes 16-31
    const int feat = lane & 15;     // feature index inside fragment
    const int blk  = blockIdx.x;
    const int b    = blockIdx.y;

    const float FOCAL = 608.365f;
    const float CXf = 256.0f, CYf = 256.0f;
    const float PI  = 3.14159f;                       // matches reference exactly
    const float c0 = PI * sqrtf(1.0f / (4.0f * PI));                // folded
    const float c1 = (2.0f * PI / 3.0f) * sqrtf(3.0f / (4.0f * PI));
    const float c2 = (PI / 4.0f) * 0.5f * sqrtf(5.0f / (4.0f * PI));
    const float c3 = (PI / 4.0f) * 3.0f * sqrtf(5.0f / (12.0f * PI));
    const float c4 = (PI / 4.0f) * 3.0f * sqrtf(5.0f / (48.0f * PI));
    const float invF = 1.0f / FOCAL;

    const float* dep  = depth_target + (size_t)b * HW_;
    const float* img  = input_img   + (size_t)b * 3 * HW_;
    const float* alb2 = albedo      + (size_t)b * 3 * HW_ + 2 * (size_t)HW_;

    v8f acc = {};   // 16x16 f32 Gram accumulator fragment (8 VGPRs)

    for (int it = 0; it < ITERS; ++it) {
        const int p = blk * PIX_PER_BLOCK + it * 256 + tid;
        const int r = p >> 9;           // W_ == 512
        const int c = p & (W_ - 1);

        // clamped neighbor indices (values discarded at border)
        const int cm = (c > 0)      ? c - 1 : c;
        const int cp = (c < W_ - 1) ? c + 1 : c;
        const int rm = (r > 0)      ? r - 1 : r;
        const int rp = (r < H_ - 1) ? r + 1 : r;

        const float dC = (dep[r  * W_ + c ] + 1.0f) * 0.5f;
        const float dE = (dep[r  * W_ + cp] + 1.0f) * 0.5f;
        const float dW = (dep[r  * W_ + cm] + 1.0f) * 0.5f;
        const float dS = (dep[rp * W_ + c ] + 1.0f) * 0.5f;
        const float dN = (dep[rm * W_ + c ] + 1.0f) * 0.5f;

        const float xc = ((float)c - CXf) * invF;
        const float yc = ((float)r - CYf) * invF;

        // back-projected points (x,y,z)
        const float Pcx = xc * dC,            Pcy = yc * dC,            Pcz = dC;
        const float Pex = (xc + invF) * dE,   Pey = yc * dE,            Pez = dE;
        const float Pwx = (xc - invF) * dW,   Pwy = yc * dW,            Pwz = dW;
        const float Psx = xc * dS,            Psy = (yc + invF) * dS,   Psz = dS;
        const float Pnx = xc * dN,            Pny = (yc - invF) * dN,   Pnz = dN;

        // vw = C - E ; vs = S - C ; ve = C - W ; vn = N - C
        const float vwx = Pcx - Pex, vwy = Pcy - Pey, vwz = Pcz - Pez;
        const float vsx = Psx - Pcx, vsy = Psy - Pcy, vsz = Psz - Pcz;
        const float vex = Pcx - Pwx, vey = Pcy - Pwy, vez = Pcz - Pwz;
        const float vnx = Pnx - Pcx, vny = Pny - Pcy, vnz = Pnz - Pcz;

        // normal = cross(vw, vs) + cross(ve, vn)
        float nx = (vwy * vsz - vwz * vsy) + (vey * vnz - vez * vny);
        float ny = (vwz * vsx - vwx * vsz) + (vez * vnx - vex * vnz);
        float nz = (vwx * vsy - vwy * vsx) + (vex * vny - vey * vnx);

        const bool interior = (r > 0) & (r < H_ - 1) & (c > 0) & (c < W_ - 1);
        nx = interior ? nx : 0.0f;
        ny = interior ? ny : 0.0f;
        nz = interior ? nz : 0.0f;

        const float mag = sqrtf(nx * nx + ny * ny + nz * nz);
        const float inv = (mag > 0.0f) ? (1.0f / mag) : 0.0f;

        // image channels, mask, albedo
        const float i0 = (img[0 * HW_ + p] + 1.0f) * 0.5f;
        const float i1 = (img[1 * HW_ + p] + 1.0f) * 0.5f;
        const float i2 = (img[2 * HW_ + p] + 1.0f) * 0.5f;
        const float m  = (i0 != 0.0f) ? 1.0f : 0.0f;
        const float a2 = alb2[p] * m;

        // SH basis * (albedo2 * mask); quadratic terms use unit normal:
        // (u_i u_j) * mag == n_i n_j / mag
        float G[16];
        G[0] = c0 * mag * a2;
        G[1] = c1 * nz * a2;
        G[2] = c1 * nx * a2;
        G[3] = c1 * ny * a2;
        G[4] = c2 * (2.0f * nz * nz - nx * nx - ny * ny) * inv * a2;
        G[5] = c3 * nx * nz * inv * a2;
        G[6] = c3 * ny * nz * inv * a2;
        G[7] = c4 * (nx * nx - ny * ny) * inv * a2;
        G[8] = c3 * nx * ny * inv * a2;
        G[9]  = i0 * m;
        G[10] = i1 * m;
        G[11] = i2 * m;
        G[12] = 0.0f; G[13] = 0.0f; G[14] = 0.0f; G[15] = 0.0f;

        const int base = tid * LDS_STRIDE;
        #pragma unroll
        for (int i = 0; i < 16; ++i) sh[base + i] = G[i];

        __syncthreads();

        // Wave-private 32x16 sub-tile -> 8 WMMAs, 4 pixels (K=4) each.
        const int wbase = wave * 32;
        #pragma unroll
        for (int j = 0; j < 8; ++j) {
            const int q = wbase + j * 4 + 2 * half;   // K0|K2 row for this lane half
            v2f fa;
            fa.x = sh[q * LDS_STRIDE + feat];         // K = 0 (or 2)
            fa.y = sh[(q + 1) * LDS_STRIDE + feat];   // K = 1 (or 3)
            // D = A * B + C ; identical register data serves A and its transpose B
            acc = __builtin_amdgcn_wmma_f32_16x16x4_f32(
                false, fa, false, fa, (short)0, acc, false, false);
        }
        __syncthreads();   // protect LDS reuse by next iteration
    }

    // C/D layout: VGPR v, lanes 0-15 -> (M=v,   N=lane),
    //                     lanes16-31 -> (M=v+8, N=lane-16)
    #pragma unroll
    for (int v = 0; v < 8; ++v) {
        const int M = v + 8 * half;
        red[wave * 256 + M * 16 + feat] = acc[v];
    }
    __syncthreads();

    float s = 0.0f;
    #pragma unroll
    for (int w = 0; w < 8; ++w) s += red[w * 256 + tid];
    partial[((size_t)b * BLOCKS_PER_BATCH + blk) * 256 + tid] = s;
}

// ---------------------------------------------------------------------------
// Kernel 2: reduce per-block Gram partials (deterministic order) and solve
// AtA * x = Atb for 3 RHS per batch (Gauss-Jordan, partial pivoting).
// ---------------------------------------------------------------------------
__global__ __launch_bounds__(256)
void sh_solve_kernel(const float* __restrict__ partial, float* __restrict__ out)
{
    __shared__ float gram[256];
    const int b = blockIdx.x;
    const int t = threadIdx.x;

    float s = 0.0f;
    for (int blk = 0; blk < BLOCKS_PER_BATCH; ++blk)
        s += partial[((size_t)b * BLOCKS_PER_BATCH + blk) * 256 + t];
    gram[t] = s;
    __syncthreads();

    if (t == 0) {
        float M[9][12];
        for (int i = 0; i < 9; ++i) {
            for (int j = 0; j < 9; ++j) M[i][j] = gram[i * 16 + j];      // AtA
            for (int ch = 0; ch < 3; ++ch) M[i][9 + ch] = gram[(9 + ch) * 16 + i]; // Atb
        }
        for (int k = 0; k < 9; ++k) {
            int piv = k;
            float best = fabsf(M[k][k]);
            for (int rr = k + 1; rr < 9; ++rr) {
                const float v = fabsf(M[rr][k]);
                if (v > best) { best = v; piv = rr; }
            }
            if (piv != k) {
                for (int j = 0; j < 12; ++j) {
                    const float tmp = M[k][j]; M[k][j] = M[piv][j]; M[piv][j] = tmp;
                }
            }
            const float invp = 1.0f / M[k][k];
            for (int j = 0; j < 12; ++j) M[k][j] *= invp;
            for (int rr = 0; rr < 9; ++rr) {
                if (rr == k) continue;
                const float f = M[rr][k];
                for (int j = 0; j < 12; ++j) M[rr][j] -= f * M[k][j];
            }
        }
        for (int ch = 0; ch < 3; ++ch)
            for (int i = 0; i < 9; ++i)
                out[b * 27 + ch * 9 + i] = M[i][9 + ch];
    }
}

extern "C" void kernel_launch(void* const* d_in, const int* in_sizes, int n_in,
                              void* d_out, int out_size, void* d_ws, size_t ws_size,
                              hipStream_t stream)
{
    const float* input_img    = (const float*)d_in[0];
    const float* depth_target = (const float*)d_in[1];
    const float* albedo       = (const float*)d_in[2];
    float* partial = (float*)d_ws;   // NBATCH * BLOCKS_PER_BATCH * 256 floats = 512 KB

    dim3 grid(BLOCKS_PER_BATCH, NBATCH);
    sh_gram_kernel<<<grid, 256, 0, stream>>>(input_img, depth_target, albedo, partial);
    sh_solve_kernel<<<NBATCH, 256, 0, stream>>>(partial, (float*)d_out);
}